// DigitCaps_24292335026665
// MI455X (gfx1250) — compile-verified
//
#include <hip/hip_runtime.h>

// CDNA5 / gfx1250, wave32.
// Reference reduces (see analysis) to: per-row mean-scaled sum -> squash -> 16-group L2 norm.
// HBM-bound: 590MB streamed once; NT loads since footprint > 192MB L2.

typedef float v2f __attribute__((ext_vector_type(2)));
typedef float v4f __attribute__((ext_vector_type(4)));
typedef float v8f __attribute__((ext_vector_type(8)));

#define D_COLS 9216           // 1152 * 8
#define ROWS_PER_BLOCK 16     // one output element per block
#define VEC_PER_LANE (D_COLS / (32 * 4))   // 72 float4 loads per lane

__global__ __launch_bounds__(512) void digitcaps_kernel(const float* __restrict__ x,
                                                        float* __restrict__ out,
                                                        int n_out) {
    const int lane = threadIdx.x & 31;
    const int wave = threadIdx.x >> 5;          // 0..15, one row per wave
    const int row  = blockIdx.x * ROWS_PER_BLOCK + wave;

    // ---- Streaming row sum: consecutive lanes -> consecutive float4 (512B per wave op)
    const v4f* vp = (const v4f*)(x + (size_t)row * D_COLS);
    v4f acc = {0.f, 0.f, 0.f, 0.f};
#pragma unroll 8
    for (int k = 0; k < VEC_PER_LANE; ++k) {
        v4f v = __builtin_nontemporal_load(&vp[k * 32 + lane]);   // global_load_b128, TH=NT
        acc += v;
    }
    float p = (acc.x + acc.y) + (acc.z + acc.w);  // per-lane partial

    // ---- Cross-lane reduce, stage 1 via WMMA (f32, no precision change):
    // A = ones(16x4), B holds the 32 partials in one VGPR (other VGPR zero).
    // => D[m,n] = column-sum of B, identical for every row m, replicated in both
    //    16-lane halves of the C/D layout. Sum of all 16 columns == sum of all 32 partials.
    v2f a = {1.0f, 1.0f};   // all-ones A matrix
    v2f b = {p, 0.0f};      // partials occupy exactly one B VGPR
    v8f c = {};
    c = __builtin_amdgcn_wmma_f32_16x16x4_f32(
            /*neg_a=*/false, a, /*neg_b=*/false, b,
            /*c_mod=*/(short)0, c, /*reuse_a=*/false, /*reuse_b=*/false);
    float r = c[0];         // lane L holds colsum(L & 15)

    // ---- Stage 2: sum the 16 column-sums (data already mirrored across halves)
    r += __shfl_xor(r, 8, 32);
    r += __shfl_xor(r, 4, 32);
    r += __shfl_xor(r, 2, 32);
    r += __shfl_xor(r, 1, 32);

    // ---- Per-row epilogue: t = sum * softmax-uniform (1/9216); |squash(t)|^2 = (t^2/(1+t^2))^2
    __shared__ float cross[ROWS_PER_BLOCK];
    if (lane == 0) {
        float t  = r * (1.0f / 9216.0f);
        float t2 = t * t;
        float g  = t2 / (1.0f + t2);
        cross[wave] = g * g;
    }
    __syncthreads();

    // ---- Final 16-group L2 norm -> one output per block
    if (threadIdx.x == 0) {
        float s = 0.f;
#pragma unroll
        for (int i = 0; i < ROWS_PER_BLOCK; ++i) s += cross[i];
        if (blockIdx.x < (unsigned)n_out) out[blockIdx.x] = sqrtf(s);
    }
}

extern "C" void kernel_launch(void* const* d_in, const int* in_sizes, int n_in,
                              void* d_out, int out_size, void* d_ws, size_t ws_size,
                              hipStream_t stream) {
    (void)n_in; (void)d_ws; (void)ws_size;
    const float* x = (const float*)d_in[0];
    float* out = (float*)d_out;
    // in_sizes[0] = B * 9216; out_size = B/16 (= number of blocks, 16 rows each)
    int nblocks = out_size;  // 1000 for the reference shapes
    digitcaps_kernel<<<nblocks, 512, 0, stream>>>(x, out, out_size);
}